// RGCN_60026462929566
// MI455X (gfx1250) — compile-verified
//
#include <hip/hip_runtime.h>
#include <hip/hip_bf16.h>

// ---------------------------------------------------------------------------
// RGCN on MI455X (gfx1250, wave32, WMMA)
// Stage 1: hw[n][r][o] = sum_i h[n][i] * W[r][i][o]   via V_WMMA_F32_16X16X4_F32
// Stage 2: out[dst[e]] += hw[src[e]][rel[e]]          via global_atomic_add_f32
// ---------------------------------------------------------------------------

#define IN_FEAT  128
#define OUT_FEAT 128
#define NUM_REL  8

// LDS holds W[r] transposed: wlds[o][i], padded stride so that B-fragment
// ds_load_b64 reads (addr = (o*LDS_STRIDE + k)*4) hit distinct banks:
// bank = (o*132 + k) % 64 = (4*o + k) % 64  -> conflict-free across the wave.
#define LDS_STRIDE 132   // floats; 128 * 132 * 4 = 67,584 bytes (< 320KB/WGP)

typedef __attribute__((ext_vector_type(2))) float v2f;
typedef __attribute__((ext_vector_type(8))) float v8f;

// ---------------------------------------------------------------------------
// Stage 1: batched GEMM, one wave computes a [16 nodes x 128 out] tile for one
// relation. 4 waves (128 threads) per block share the LDS copy of W[r]^T.
// ---------------------------------------------------------------------------
__global__ __launch_bounds__(128) void rgcn_gemm(const float* __restrict__ h,
                                                 const float* __restrict__ w,
                                                 float* __restrict__ hw,
                                                 int n_nodes) {
    __shared__ float wlds[OUT_FEAT * LDS_STRIDE];

    const int r = blockIdx.y;
    const float* __restrict__ wr = w + (size_t)r * IN_FEAT * OUT_FEAT;

    // Stage W[r] transposed into LDS: wlds[o*LDS_STRIDE + i] = W[r][i][o].
    // Global reads fully coalesced (idx is linear over [i][o]).
    for (int idx = threadIdx.x; idx < IN_FEAT * OUT_FEAT; idx += 128) {
        const int i = idx >> 7;    // in-feature (K)
        const int o = idx & 127;   // out-feature (N)
        wlds[o * LDS_STRIDE + i] = wr[idx];
    }
    __syncthreads();

    const int wave = threadIdx.x >> 5;              // 0..3
    const int lane = threadIdx.x & 31;
    const int tile = blockIdx.x * 4 + wave;         // 16-node tile index
    const int m0   = tile * 16;
    if (m0 >= n_nodes) return;                      // wave-uniform: EXEC stays all-1s

    const int mrow = lane & 15;                     // M (A) / N (B,C) position
    const int khi  = (lane >> 4) << 1;              // lanes 16..31 hold K+2,K+3

    // A-matrix 16x4 fp32 layout (ISA 7.12.2): lane = M row, VGPR0/1 = K pair,
    // lane/16 selects which K pair -> per k-step each lane loads 2 adjacent floats.
    const float* __restrict__ aptr = h + (size_t)(m0 + mrow) * IN_FEAT + khi;

    v8f acc[8] = {};   // 8 tiles of 16x16 f32 accumulators (N = 0..127)

    for (int k = 0; k < IN_FEAT; k += 4) {
        const v2f a = *(const v2f*)(aptr + k);
#pragma unroll
        for (int t8 = 0; t8 < 8; ++t8) {
            const int o = t8 * 16 + mrow;
            // B-matrix 4x16: VGPR0 = row K (lanes 0-15) / K+2 (lanes 16-31),
            // VGPR1 = K+1 / K+3. Transposed-LDS makes this one b64 load.
            const v2f b = *(const v2f*)(&wlds[o * LDS_STRIDE + k + khi]);
            acc[t8] = __builtin_amdgcn_wmma_f32_16x16x4_f32(
                /*neg_a=*/false, a, /*neg_b=*/false, b,
                /*c_mod=*/(short)0, acc[t8],
                /*reuse_a=*/false, /*reuse_b=*/false);
        }
    }

    // C/D layout: VGPR v -> M = v (lanes 0-15) or v+8 (lanes 16-31), N = lane%16.
    const int mbase = (lane < 16) ? 0 : 8;
#pragma unroll
    for (int t8 = 0; t8 < 8; ++t8) {
        const int o = t8 * 16 + mrow;
#pragma unroll
        for (int v = 0; v < 8; ++v) {
            const int m = mbase + v;
            hw[((size_t)(m0 + m) * NUM_REL + r) * OUT_FEAT + o] = acc[t8][v];
        }
    }
}

// ---------------------------------------------------------------------------
// Stage 2: one wave per edge; lane L handles out features 4L..4L+3.
// ---------------------------------------------------------------------------
__global__ __launch_bounds__(256) void rgcn_agg(const float* __restrict__ hw,
                                                const int* __restrict__ src,
                                                const int* __restrict__ dst,
                                                const int* __restrict__ rel,
                                                float* __restrict__ out,
                                                int n_edges) {
    const int gid  = blockIdx.x * blockDim.x + threadIdx.x;
    const int e    = gid >> 5;
    const int lane = gid & 31;
    if (e >= n_edges) return;

    const int s = src[e];
    const int d = dst[e];
    const int r = rel[e];

    const float4 v = *(const float4*)(hw + ((size_t)s * NUM_REL + r) * OUT_FEAT + lane * 4);
    float* op = out + (size_t)d * OUT_FEAT + lane * 4;
    atomicAdd(op + 0, v.x);
    atomicAdd(op + 1, v.y);
    atomicAdd(op + 2, v.z);
    atomicAdd(op + 3, v.w);
}

__global__ void rgcn_zero(float* __restrict__ out, int n) {
    const int i = blockIdx.x * blockDim.x + threadIdx.x;
    if (i < n) out[i] = 0.0f;
}

// ---------------------------------------------------------------------------
extern "C" void kernel_launch(void* const* d_in, const int* in_sizes, int n_in,
                              void* d_out, int out_size, void* d_ws, size_t ws_size,
                              hipStream_t stream) {
    const float* h   = (const float*)d_in[0];   // [N, 128] fp32
    const float* w   = (const float*)d_in[1];   // [8, 128, 128] fp32
    const int*   src = (const int*)d_in[2];     // [E]
    const int*   dst = (const int*)d_in[3];     // [E]
    const int*   rel = (const int*)d_in[4];     // [E]
    float*       out = (float*)d_out;           // [N, 128] fp32
    float*       hw  = (float*)d_ws;            // [N, 8, 128] fp32 scratch (204.8 MB)

    const int n_nodes = in_sizes[0] / IN_FEAT;
    const int n_edges = in_sizes[2];

    // Zero the (poisoned) output before atomic accumulation.
    rgcn_zero<<<(out_size + 255) / 256, 256, 0, stream>>>(out, out_size);

    // Stage 1: 4 node-tiles of 16 per block, one relation per blockIdx.y.
    const int tiles = (n_nodes + 15) / 16;
    dim3 g1((tiles + 3) / 4, NUM_REL);
    rgcn_gemm<<<g1, 128, 0, stream>>>(h, w, hw, n_nodes);

    // Stage 2: one wave (32 lanes) per edge.
    const long long total_threads = (long long)n_edges * 32;
    const int blocks = (int)((total_threads + 255) / 256);
    rgcn_agg<<<blocks, 256, 0, stream>>>(hw, src, dst, rel, out, n_edges);
}